// MarginContrastiveLoss_34462817583460
// MI455X (gfx1250) — compile-verified
//
#include <hip/hip_runtime.h>
#include <hip/hip_bf16.h>

#define BSZ 512
#define DDIM 256
#define MARGIN 0.2f

typedef __attribute__((ext_vector_type(2))) float v2f;
typedef __attribute__((ext_vector_type(8))) float v8f;

// ---------------------------------------------------------------------------
// Kernel 1: sim = F @ F^T  with V_WMMA_F32_16X16X4_F32 (exact fp32 path).
// One wave per 16x16 tile of sim. Grid = (32, 32), block = 32 (one wave32).
//
// A-fragment layout (32-bit A 16x4, ISA 7.12.2):
//   v0: lanes 0-15 -> (M=lane, K=0), lanes 16-31 -> (M=lane-16, K=2)
//   v1: lanes 0-15 -> (M=lane, K=1), lanes 16-31 -> (M=lane-16, K=3)
// B-fragment (4x16, B[k][n] = F[n][k]) mirrored with N = lane%16.
// C/D layout: VGPR r: lanes 0-15 -> (M=r, N=lane), lanes 16-31 -> (M=r+8).
// ---------------------------------------------------------------------------
__global__ __launch_bounds__(32) void sim_wmma_kernel(const float* __restrict__ F,
                                                      float* __restrict__ sim) {
    const int tm   = blockIdx.x;          // tile row   (0..31)
    const int tn   = blockIdx.y;          // tile col   (0..31)
    const int lane = threadIdx.x;         // 0..31, full wave, EXEC all ones
    const int half = lane >> 4;           // 0: K pair {0,1}, 1: K pair {2,3}
    const int l    = lane & 15;

    const float* __restrict__ Arow = F + (size_t)(tm * 16 + l) * DDIM;  // A row M=l
    const float* __restrict__ Bcol = F + (size_t)(tn * 16 + l) * DDIM;  // B col N=l

    v8f acc = {};
    #pragma unroll 4
    for (int k = 0; k < DDIM; k += 4) {
        v2f a, b;
        a.x = Arow[k + 2 * half + 0];
        a.y = Arow[k + 2 * half + 1];
        b.x = Bcol[k + 2 * half + 0];
        b.y = Bcol[k + 2 * half + 1];
        // 8 args: (neg_a, A, neg_b, B, c_mod, C, reuse_a, reuse_b)
        acc = __builtin_amdgcn_wmma_f32_16x16x4_f32(false, a, false, b,
                                                    (short)0, acc, false, false);
    }

    const int n     = tn * 16 + l;
    const int mbase = tm * 16 + half * 8;
    #pragma unroll
    for (int r = 0; r < 8; ++r) {
        sim[(size_t)(mbase + r) * BSZ + n] = acc[r];
    }
}

// ---------------------------------------------------------------------------
// Kernel 2: per-row positive/negative counts + num_valid.
// cnts[0..511] = pos_cnt, cnts[512..1023] = neg_cnt, cnts[1024] = num_valid.
// One block of 512 threads; deterministic LDS tree reduction.
// ---------------------------------------------------------------------------
__global__ __launch_bounds__(BSZ) void counts_kernel(const int* __restrict__ labels,
                                                     float* __restrict__ cnts) {
    __shared__ int   sl[BSZ];
    __shared__ float red[BSZ];
    const int i = threadIdx.x;
    sl[i] = labels[i];
    __syncthreads();

    const int my = sl[i];
    int eq = 0;
    #pragma unroll 8
    for (int j = 0; j < BSZ; ++j) eq += (sl[j] == my) ? 1 : 0;

    const float pos = (float)(eq - 1);      // diagonal removed
    const float neg = (float)(BSZ - eq);
    cnts[i]       = pos;
    cnts[BSZ + i] = neg;

    red[i] = (pos > 0.0f && neg > 0.0f) ? 1.0f : 0.0f;
    __syncthreads();
    for (int s = BSZ / 2; s > 0; s >>= 1) {
        if (i < s) red[i] += red[i + s];
        __syncthreads();
    }
    if (i == 0) cnts[2 * BSZ] = red[0];     // num_valid
}

// ---------------------------------------------------------------------------
// Kernel 3: per-anchor hinge sum.
//   rowsum[i] = sum_{j in pos(i)} sum_{k in neg(i)} relu(m + s_ik - s_ij)
// One block per anchor row; sim row + labels staged in LDS.
// ---------------------------------------------------------------------------
__global__ __launch_bounds__(256) void hinge_kernel(const float* __restrict__ sim,
                                                    const int* __restrict__ labels,
                                                    float* __restrict__ rowsum) {
    __shared__ float srow[BSZ];
    __shared__ int   sl[BSZ];
    __shared__ float red[256];

    const int i = blockIdx.x;
    const int t = threadIdx.x;

    for (int j = t; j < BSZ; j += 256) {
        srow[j] = sim[(size_t)i * BSZ + j];
        sl[j]   = labels[j];
    }
    __syncthreads();

    const int li = sl[i];
    float acc = 0.0f;
    for (int j = t; j < BSZ; j += 256) {
        if (j == i || sl[j] != li) continue;          // pos_mask: same label, off-diag
        const float base = MARGIN - srow[j];
        for (int k = 0; k < BSZ; ++k) {
            if (sl[k] == li) continue;                // neg_mask: different label
            const float h = base + srow[k];
            acc += (h > 0.0f) ? h : 0.0f;
        }
    }

    red[t] = acc;
    __syncthreads();
    for (int s = 128; s > 0; s >>= 1) {
        if (t < s) red[t] += red[t + s];
        __syncthreads();
    }
    if (t == 0) rowsum[i] = red[0];
}

// ---------------------------------------------------------------------------
// Kernel 4: deterministic final sum + buggy denominator from the reference:
//   denom = num_valid * pos_cnt[B-1] * neg_cnt[B-1]
// ---------------------------------------------------------------------------
__global__ __launch_bounds__(BSZ) void finalize_kernel(const float* __restrict__ rowsum,
                                                       const float* __restrict__ cnts,
                                                       float* __restrict__ out) {
    __shared__ float red[BSZ];
    const int t = threadIdx.x;
    red[t] = rowsum[t];
    __syncthreads();
    for (int s = BSZ / 2; s > 0; s >>= 1) {
        if (t < s) red[t] += red[t + s];
        __syncthreads();
    }
    if (t == 0) {
        const float loss  = red[0];
        const float denom = cnts[2 * BSZ] * cnts[BSZ - 1] * cnts[2 * BSZ - 1];
        out[0] = (denom > 0.0f) ? (loss / denom) : 0.0f;
    }
}

// ---------------------------------------------------------------------------
extern "C" void kernel_launch(void* const* d_in, const int* in_sizes, int n_in,
                              void* d_out, int out_size, void* d_ws, size_t ws_size,
                              hipStream_t stream) {
    const float* features = (const float*)d_in[0];   // [512, 256] f32
    const int*   labels   = (const int*)d_in[1];     // [512] i32
    float*       out      = (float*)d_out;           // scalar f32

    float* sim    = (float*)d_ws;                    // 512*512 f32 = 1 MB
    float* cnts   = sim + (size_t)BSZ * BSZ;         // 2*512 + 1 f32
    float* rowsum = cnts + 2 * BSZ + 64;             // 512 f32 (padded offset)

    dim3 gGemm(BSZ / 16, BSZ / 16);                  // 32x32 tiles
    sim_wmma_kernel<<<gGemm, 32, 0, stream>>>(features, sim);
    counts_kernel<<<1, BSZ, 0, stream>>>(labels, cnts);
    hinge_kernel<<<BSZ, 256, 0, stream>>>(sim, labels, rowsum);
    finalize_kernel<<<1, BSZ, 0, stream>>>(rowsum, cnts, out);
}